// PINNPhysicsLoss_4277787427055
// MI455X (gfx1250) — compile-verified
//
#include <hip/hip_runtime.h>

#define PP 7
#define BRIGHT 0.02f
#define BIGV 1000000000.0f
#define KNBR 5
#define BATCH 16
#define NPTS 2000
#define HIMG 720
#define WIMG 1280
#define BN (BATCH * NPTS)

typedef float v2f __attribute__((ext_vector_type(2)));
typedef float v8f __attribute__((ext_vector_type(8)));

// Workspace layout (floats):
//   [0..6]    global scalars: Ssc, Syd_s, Syd, Wsum, Sdiff_w, Ssl1_w, cnt(sc>0.1)
//   [32..47]  n_valid[b]
//   [48..63]  hsum[b]
//   [64..79]  ls_sum[b]
//   [80..95]  pen_sum[b]
//   [128 ...] pts4[BN] (float4: x, z, h, sq), then vs[BN]
#define WS_PTS 128

__device__ __forceinline__ void wavebar() {
#if __has_builtin(__builtin_amdgcn_wave_barrier)
    __builtin_amdgcn_wave_barrier();   // code-motion fence; HW DS ops are in-order per wave
#else
    __syncthreads();
#endif
}

__device__ __forceinline__ float smooth_l1(float x, float y, float beta) {
    float d = fabsf(x - y);
    return (d < beta) ? (0.5f * d * d / beta) : (d - 0.5f * beta);
}

__device__ __forceinline__ float bilin(const float* im, float kx, float ky,
                                       float ox, float oy) {
    float px = fminf(fmaxf(kx + ox, 0.0f), (float)(WIMG - 1));
    float py = fminf(fmaxf(ky + oy, 0.0f), (float)(HIMG - 1));
    float x0 = floorf(px), y0 = floorf(py);
    float fx = px - x0, fy = py - y0;
    int x0i = min(max((int)x0, 0), WIMG - 1);
    int x1i = min(x0i + 1, WIMG - 1);
    int y0i = min(max((int)y0, 0), HIMG - 1);
    int y1i = min(y0i + 1, HIMG - 1);
    float v00 = im[y0i * WIMG + x0i];
    float v01 = im[y0i * WIMG + x1i];
    float v10 = im[y1i * WIMG + x0i];
    float v11 = im[y1i * WIMG + x1i];
    return (v00 * (1.0f - fx) + v01 * fx) * (1.0f - fy) +
           (v10 * (1.0f - fx) + v11 * fx) * fy;
}

__global__ void k_init(float* ws) {
    int i = threadIdx.x;
    if (i < 128) ws[i] = 0.0f;
}

// One thread per keypoint: epipolar, patches, photo terms, projection, features.
__global__ void k_point(const float* __restrict__ lg, const float* __restrict__ rg,
                        const float* __restrict__ kpl, const float* __restrict__ kpr,
                        const float* __restrict__ scores, const float* __restrict__ Q,
                        float* ws) {
    int gid = blockIdx.x * blockDim.x + threadIdx.x;   // grid sized exactly BN
    int b = gid / NPTS;

    float xl = kpl[gid * 2 + 0], yl = kpl[gid * 2 + 1];
    float xr = kpr[gid * 2 + 0], yr = kpr[gid * 2 + 1];
    float sc = scores[gid];
    float yd = fabsf(yl - yr);

    const float* iml = lg + (size_t)b * HIMG * WIMG;
    const float* imr = rg + (size_t)b * HIMG * WIMG;

    float sum_ad = 0.0f, cl = 0.0f, cr = 0.0f;
    for (int t = 0; t < PP * PP; t++) {
        float ox = (float)(t % PP - 3);
        float oy = (float)(t / PP - 3);
        float a = bilin(iml, xl, yl, ox, oy);
        float c = bilin(imr, xr, yr, ox, oy);
        sum_ad += fabsf(a - c);
        if (t == 24) { cl = a; cr = c; }
    }
    float diff = sum_ad / 49.0f;
    float isb = (cl > BRIGHT) ? 1.0f : 0.0f;
    float w = sc * isb;
    float sl1 = smooth_l1(cl, cr, 1.0f);

    // projection: pts4 = [xl, yl, disp, 1];  proj_j = sum_i pts4_i * Q[b][j][i]
    float disp = xl - xr;
    const float* Qb = Q + b * 16;
    float p0 = xl * Qb[0]  + yl * Qb[1]  + disp * Qb[2]  + Qb[3];
    float p1 = xl * Qb[4]  + yl * Qb[5]  + disp * Qb[6]  + Qb[7];
    float p2 = xl * Qb[8]  + yl * Qb[9]  + disp * Qb[10] + Qb[11];
    float p3 = xl * Qb[12] + yl * Qb[13] + disp * Qb[14] + Qb[15];
    float Wc = fmaxf(p3, 1e-6f);
    float X = p0 / Wc, Y = p1 / Wc, Z = p2 / Wc;
    bool valid = (Z > 100.0f) && (Z < 30000.0f) && (sc > 0.1f);
    float xm = X / 1000.0f, hm = Y / 1000.0f, zm = Z / 1000.0f;
    float sq = xm * xm + zm * zm;

    float4* pts4 = (float4*)(ws + WS_PTS);
    float* vs = ws + WS_PTS + 4 * BN;
    pts4[gid] = make_float4(xm, zm, hm, sq);
    vs[gid] = valid ? 1.0f : 0.0f;

    if (valid) {
        atomicAdd(&ws[32 + b], 1.0f);
        atomicAdd(&ws[48 + b], hm);
    }

    float g[7];
    g[0] = sc;
    g[1] = yd * sc;
    g[2] = yd;
    g[3] = w;
    g[4] = diff * w;
    g[5] = sl1 * w;
    g[6] = (sc > 0.1f) ? 1.0f : 0.0f;
#pragma unroll
    for (int k = 0; k < 7; k++) {
        float v = g[k];
        for (int o = 16; o > 0; o >>= 1) v += __shfl_down(v, o, 32);
        if ((threadIdx.x & 31) == 0) atomicAdd(&ws[k], v);
    }
}

// KNN + neighbor losses. One block = 8 independent waves; each wave owns a
// 16-row tile and a private LDS tile buffer (no block barriers in the loop:
// same-wave DS ops complete in order, wavebar() stops compiler reordering).
// d2 tile (16x16) = A(16x4) x B(4x16) via V_WMMA_F32_16X16X4_F32 where
// A row n = [x_n, z_n, sq_n, 1], B col m = [-2x_m, -2z_m, 1, sq_m].
__global__ void k_knn(float* ws) {
    // single flat LDS array -> ternary selects become address cndmask, not branches
    __shared__ float smem[5 * NPTS + 8 * 256];
    const int OX = 0, OZ = NPTS, OQ = 2 * NPTS, OH = 3 * NPTS, OV = 4 * NPTS,
              OT = 5 * NPTS;

    int b = blockIdx.y;
    const float4* pts4 = (const float4*)(ws + WS_PTS);
    const float* vsA = ws + WS_PTS + 4 * BN;

    int tid = threadIdx.x;
    for (int i = tid; i < NPTS; i += 256) {
        float4 p = pts4[b * NPTS + i];
        smem[OX + i] = p.x;
        smem[OZ + i] = p.y;
        smem[OH + i] = p.z;
        smem[OQ + i] = p.w;
        smem[OV + i] = vsA[b * NPTS + i];
    }
    __syncthreads();   // features shared across waves: one real barrier

    int wave = tid >> 5;
    int l = tid & 31;
    int li = l & 15;
    int hi = l >> 4;
    int rowTile = blockIdx.x * 8 + wave;          // 0..127; active < 125
    bool tileAct = rowTile < (NPTS / 16);
    int gr = rowTile * 16 + li;
    int grc = min(gr, NPTS - 1);
    float scaleB = hi ? 1.0f : -2.0f;

    // A operand: lanes 0-15 hold K0/K1, lanes 16-31 hold K2/K3 (M = lane%16)
    v2f a;
    a.x = smem[(hi ? OQ : OX) + grc];             // sq_n : x_n  (address select)
    {
        float zv = smem[OZ + grc];
        a.y = hi ? 1.0f : zv;
    }
    bool rowValid = tileAct && (smem[OV + grc] > 0.5f);
    float myh = smem[OH + grc];

    float tv[6];
    int ti[6];
#pragma unroll
    for (int k = 0; k < 6; k++) { tv[k] = 3.4e38f; ti[k] = 0; }

    float* tile = &smem[OT + wave * 256];

    for (int ct = 0; ct < NPTS / 16; ct++) {
        int cb = ct * 16;
        int gm = cb + li;
        // B operand: col m = [-2x, -2z, 1, sq]; branchless via address select + scale
        float xv = smem[OX + gm];
        float ov = smem[(hi ? OQ : OZ) + gm];     // sq_m : z_m
        v2f bb;
        bb.x = hi ? 1.0f : (-2.0f * xv);
        bb.y = ov * scaleB;                        // hi: sq_m, lo: -2*z_m
        v8f c = {0.f, 0.f, 0.f, 0.f, 0.f, 0.f, 0.f, 0.f};
#if __has_builtin(__builtin_amdgcn_wmma_f32_16x16x4_f32)
        c = __builtin_amdgcn_wmma_f32_16x16x4_f32(
            false, a, false, bb, (short)0, c, false, false);
#else
#pragma unroll
        for (int r = 0; r < 8; r++) {
            int row = r + 8 * hi;
            int grr = min(rowTile * 16 + row, NPTS - 1);
            c[r] = smem[OQ + grr] + smem[OQ + gm] -
                   2.0f * (smem[OX + grr] * xv + smem[OZ + grr] * smem[OZ + gm]);
        }
#endif
        // C/D layout: VGPR r -> row r + 8*hi, col = lane%16
#pragma unroll
        for (int r = 0; r < 8; r++) {
            int row = r + 8 * hi;
            tile[row * 16 + li] = c[r];
        }
        wavebar();   // intra-wave: DS in-order; just block compiler reordering

        // All 32 lanes scan: lane pair (l, l+16) owns row l&15, cols split 0-7 / 8-15.
        if (rowValid) {
            const float4* rowp = (const float4*)&tile[li * 16 + hi * 8];
            float4 q0 = rowp[0];
            float4 q1 = rowp[1];
            float cand[8] = {q0.x, q0.y, q0.z, q0.w, q1.x, q1.y, q1.z, q1.w};
#pragma unroll
            for (int j = 0; j < 8; j++) {
                int m = cb + hi * 8 + j;
                float dv = (smem[OV + m] > 0.5f) ? sqrtf(fmaxf(cand[j], 1e-12f))
                                                 : BIGV;
                if (dv < tv[5]) {      // per-lane stream is index-monotone: strict <
                    tv[5] = dv; ti[5] = m;
#pragma unroll
                    for (int k = 5; k > 0; k--) {
                        if (tv[k] < tv[k - 1]) {
                            float t = tv[k]; tv[k] = tv[k - 1]; tv[k - 1] = t;
                            int s = ti[k]; ti[k] = ti[k - 1]; ti[k - 1] = s;
                        }
                    }
                }
            }
        }
        wavebar();
    }

    // Merge the two half-lists of each lane pair with full lexicographic
    // (value, index) order -> exact jax.lax.top_k stable tie-break.
#pragma unroll
    for (int k = 0; k < 6; k++) {
        float pv = __shfl(tv[k], (l + 16) & 31, 32);
        int pi = __shfl(ti[k], (l + 16) & 31, 32);
        if (l < 16) {
            if (pv < tv[5] || (pv == tv[5] && pi < ti[5])) {
                tv[5] = pv; ti[5] = pi;
#pragma unroll
                for (int kk = 5; kk > 0; kk--) {
                    bool sw = (tv[kk] < tv[kk - 1]) ||
                              (tv[kk] == tv[kk - 1] && ti[kk] < ti[kk - 1]);
                    if (sw) {
                        float t = tv[kk]; tv[kk] = tv[kk - 1]; tv[kk - 1] = t;
                        int s = ti[kk]; ti[kk] = ti[kk - 1]; ti[kk - 1] = s;
                    }
                }
            }
        }
    }

    if (l < 16 && rowValid) {
        // entry 0 is self (dist ~1e-6); neighbors are entries 1..5
        float lm = (smem[OH + ti[1]] + smem[OH + ti[2]] + smem[OH + ti[3]] +
                    smem[OH + ti[4]] + smem[OH + ti[5]]) / 5.0f;
        float ls = smooth_l1(myh, lm, 0.01f);
        float pen = 0.0f;
#pragma unroll
        for (int k = 1; k < 6; k++) {
            float nd = fmaxf(tv[k], 0.001f);
            float slope = fabsf(smem[OH + ti[k]] - myh) / nd;
            pen += fmaxf(slope - 0.4f, 0.0f);
        }
        atomicAdd(&ws[64 + b], ls);
        atomicAdd(&ws[80 + b], pen);
    }
}

__global__ void k_final(const float* ws, float* out) {
    if (threadIdx.x == 0) {
        float Ss = ws[0], Syds = ws[1], Syd = ws[2];
        float Wsum = ws[3], Sdiff = ws[4], Ssl1 = ws[5], cnt = ws[6];

        float l_epi = (Ss > 1e-4f) ? (Syds / fmaxf(Ss, 1e-12f))
                                   : (Syd / (float)BN);
        float safe = fmaxf(Wsum, 1e-12f);
        float l_masked = (Wsum > 1e-4f) ? (Sdiff / safe) : 0.0f;
        float l_inten  = (Wsum > 1e-4f) ? (Ssl1 / safe) : 0.0f;
        float l_photo = l_masked + l_inten;

        float okc = 0.0f, lsS = 0.0f, lslS = 0.0f, lzS = 0.0f;
        for (int b = 0; b < BATCH; b++) {
            float nvl = ws[32 + b];
            float nv = fmaxf(nvl, 1.0f);
            float ok = (nvl >= 10.0f) ? 1.0f : 0.0f;
            float ls_b = ws[64 + b] / nv;
            float lsl_b = ws[80 + b] / (nv * (float)KNBR);
            float lz_b = fabsf(ws[48 + b] / nv);
            okc += ok;
            lsS += ls_b * ok;
            lslS += lsl_b * ok;
            lzS += lz_b * ok;
        }
        float nb = fmaxf(okc, 1.0f);
        bool gate = (cnt >= 10.0f) && (okc > 0.0f);

        out[0] = l_photo;
        out[1] = l_epi;
        out[2] = gate ? (lsS / nb) : 0.0f;
        out[3] = gate ? (lslS / nb) : 0.0f;
        out[4] = gate ? (lzS / nb) : 0.0f;
    }
}

extern "C" void kernel_launch(void* const* d_in, const int* in_sizes, int n_in,
                              void* d_out, int out_size, void* d_ws, size_t ws_size,
                              hipStream_t stream) {
    const float* lg = (const float*)d_in[0];
    const float* rg = (const float*)d_in[1];
    const float* kpl = (const float*)d_in[2];
    const float* kpr = (const float*)d_in[3];
    const float* scores = (const float*)d_in[4];
    const float* Q = (const float*)d_in[5];
    float* ws = (float*)d_ws;
    float* out = (float*)d_out;

    k_init<<<1, 128, 0, stream>>>(ws);
    k_point<<<BN / 256, 256, 0, stream>>>(lg, rg, kpl, kpr, scores, Q, ws);
    dim3 g2(16, BATCH);   // 16 row-tile blocks (8 waves x 16 rows) x 16 batches
    k_knn<<<g2, 256, 0, stream>>>(ws);
    k_final<<<1, 32, 0, stream>>>(ws, out);
}